// Model_10522669875246
// MI455X (gfx1250) — compile-verified
//
#include <hip/hip_runtime.h>

// Gated linear attention (v-axis gate), chunked WMMA formulation for gfx1250.
// B=8, S=4096, H=16, D=64. Grid = B*H*2: each WG owns one (b,h) and half of
// the value channels (state columns are independent since the gate decays
// along v). 64 sequential chunks of 64 timesteps; per chunk:
//   P  = causal(Q K^T)                          (fp32 WMMA, 64x64)
//   Y  = Gamma ⊙ (P @ (V⊘Gamma) + Q @ S0)       (64x32)
//   S0 = Gamma_last ⊙ (S0 + K^T @ (V⊘Gamma))    (64x32)
// Next chunk's Q,K,V,G are streamed with double-buffered async LDS loads.

#define Bn 8
#define Sn 4096
#define Hn 16
#define Dn 64
#define Cn 64                 // chunk (time) size
#define Vw 32                 // value columns per workgroup
#define NCHUNK (Sn / Cn)      // 64
#define LSQ 68                // LDS row stride for 64-wide tiles
#define LSV 36                // LDS row stride for 32-wide tiles
#define NTHREADS 256

typedef float v2f __attribute__((ext_vector_type(2)));
typedef float v8f __attribute__((ext_vector_type(8)));
typedef int   v4i __attribute__((ext_vector_type(4)));

#define AS1 __attribute__((address_space(1)))
#define AS3 __attribute__((address_space(3)))

#if defined(__has_builtin)
#if __has_builtin(__builtin_amdgcn_global_load_async_to_lds_b128)
#define HAS_ASYNC_LDS 1
#endif
#endif
#ifndef HAS_ASYNC_LDS
#define HAS_ASYNC_LDS 0
#endif

#if HAS_ASYNC_LDS
__device__ __forceinline__ void async_ld16(const float* gsrc, float* ldst) {
    // param0: global (AS1) v4i*, param1: LDS (AS3) v4i*, imm offset, imm cpol.
    // Route casts through integers: cross-AS pointer casts are rejected, but
    // int->pointer is fine; generic->LDS is low-32-bit truncation on amdgcn.
    __builtin_amdgcn_global_load_async_to_lds_b128(
        (AS1 v4i*)(size_t)gsrc,
        (AS3 v4i*)(unsigned int)(size_t)ldst, 0, 0);
}
#if __has_builtin(__builtin_amdgcn_s_wait_asynccnt)
#define WAIT_ASYNC(n) __builtin_amdgcn_s_wait_asynccnt(n)
#else
#define WAIT_ASYNC(n) asm volatile("s_wait_asynccnt %0" ::"n"(n) : "memory")
#endif
#endif

// A-fragment for row-major A[M][K]: lane<16 -> K pair {k0,k0+1}, lane>=16 -> {k0+2,k0+3}
template <int LST>
__device__ __forceinline__ v2f frag_row(const float* buf, int row0, int k0, int lane) {
    const int m  = lane & 15;
    const int kk = k0 + ((lane >> 4) << 1);
    v2f r;
    r.x = buf[(row0 + m) * LST + kk];
    r.y = buf[(row0 + m) * LST + kk + 1];
    return r;
}

// B-fragment for row-major B[K][N] (also A-fragment of a transposed matrix)
template <int LST>
__device__ __forceinline__ v2f frag_col(const float* buf, int k0, int col0, int lane) {
    const int n  = lane & 15;
    const int kk = k0 + ((lane >> 4) << 1);
    v2f r;
    r.x = buf[kk * LST + col0 + n];
    r.y = buf[(kk + 1) * LST + col0 + n];
    return r;
}

__device__ __forceinline__ v8f wmma_f32(v2f a, v2f b, v8f c) {
    return __builtin_amdgcn_wmma_f32_16x16x4_f32(false, a, false, b, (short)0, c,
                                                 false, false);
}

extern "C" __global__ __launch_bounds__(NTHREADS, 1)
void gla_chunk_wmma(const float* __restrict__ q, const float* __restrict__ k,
                    const float* __restrict__ v, const float* __restrict__ g,
                    float* __restrict__ out) {
    __shared__ float sQ[2][Cn * LSQ];
    __shared__ float sK[2][Cn * LSQ];
    __shared__ float sV[2][Cn * LSV];   // becomes Vhat = V / Gamma
    __shared__ float sG[2][Cn * LSV];   // becomes Gamma (inclusive cumprod in chunk)
    __shared__ float sP[Cn * LSQ];      // masked Q K^T
    __shared__ float sS[Dn * LSV];      // running state S0 [Dk x Vw]

    const int tid   = threadIdx.x;
    const int lane  = tid & 31;
    const int wid   = tid >> 5;
    const int bh    = blockIdx.x >> 1;
    const int vh    = blockIdx.x & 1;
    const int b     = bh / Hn;
    const int h     = bh % Hn;
    const int vbase = vh * Vw;

    const size_t rowstride = (size_t)Hn * Dn;                 // floats between timesteps
    const size_t base      = ((size_t)b * Sn * Hn + h) * Dn;  // offset of (b,0,h,0)

    for (int i = tid; i < Dn * LSV; i += NTHREADS) sS[i] = 0.0f;

#if HAS_ASYNC_LDS
    // ---- async copy of one chunk's Q,K (64x64) and V,G (64x32) into LDS ----
    auto issue_chunk = [&](int cc, int bsel) {
        const int t0i = cc * Cn;
        float* dQ = &sQ[bsel][0];
        float* dK = &sK[bsel][0];
        float* dV = &sV[bsel][0];
        float* dG = &sG[bsel][0];
        #pragma unroll
        for (int i = 0; i < 4; ++i) {
            const int fi = tid + i * NTHREADS, row = fi >> 4, c4 = fi & 15;
            const size_t go = base + (size_t)(t0i + row) * rowstride + (size_t)(c4 * 4);
            const int    lo = row * LSQ + c4 * 4;
            async_ld16(q + go, dQ + lo);
            async_ld16(k + go, dK + lo);
        }
        #pragma unroll
        for (int i = 0; i < 2; ++i) {
            const int fi = tid + i * NTHREADS, row = fi >> 3, c4 = fi & 7;
            const size_t go =
                base + (size_t)(t0i + row) * rowstride + (size_t)(vbase + c4 * 4);
            const int lo = row * LSV + c4 * 4;
            async_ld16(v + go, dV + lo);
            async_ld16(g + go, dG + lo);
        }
    };
    issue_chunk(0, 0);   // prime the pipeline (12 async instructions / wave)
#endif

    for (int c = 0; c < NCHUNK; ++c) {
        const int t0 = c * Cn;
        const int cb = c & 1;
        __syncthreads();   // previous chunk fully consumed before reusing its buffers

#if HAS_ASYNC_LDS
        if (c + 1 < NCHUNK) {
            issue_chunk(c + 1, cb ^ 1);  // overlap next chunk's HBM latency
            WAIT_ASYNC(12);              // in-order: chunk c's 12 loads complete
        } else {
            WAIT_ASYNC(0);
        }
#else
        // synchronous fallback
        #pragma unroll
        for (int i = 0; i < 4; ++i) {
            const int fi = tid + i * NTHREADS, row = fi >> 4, c4 = fi & 15;
            const size_t go = base + (size_t)(t0 + row) * rowstride + (size_t)(c4 * 4);
            const int    lo = row * LSQ + c4 * 4;
            *(float4*)&sQ[cb][lo] = *(const float4*)&q[go];
            *(float4*)&sK[cb][lo] = *(const float4*)&k[go];
        }
        #pragma unroll
        for (int i = 0; i < 2; ++i) {
            const int fi = tid + i * NTHREADS, row = fi >> 3, c4 = fi & 7;
            const size_t go =
                base + (size_t)(t0 + row) * rowstride + (size_t)(vbase + c4 * 4);
            const int lo = row * LSV + c4 * 4;
            *(float4*)&sV[cb][lo] = *(const float4*)&v[go];
            *(float4*)&sG[cb][lo] = *(const float4*)&g[go];
        }
#endif
        __syncthreads();

        // ---- Gamma = inclusive cumprod of G along time; Vhat = V / Gamma ----
        {
            const int col = tid & 31;   // v channel (0..31)
            const int grp = tid >> 5;   // 8 groups of 8 rows
            const int r0  = grp * 8;
            float p = 1.0f;
            #pragma unroll
            for (int r = 0; r < 8; ++r) {
                p *= sG[cb][(r0 + r) * LSV + col];
                sG[cb][(r0 + r) * LSV + col] = p;
            }
            __syncthreads();
            float pre = 1.0f;
            for (int j = 0; j < grp; ++j) pre *= sG[cb][(j * 8 + 7) * LSV + col];
            __syncthreads();   // all prefix reads before the rescale writes
            #pragma unroll
            for (int r = 0; r < 8; ++r) {
                const int off   = (r0 + r) * LSV + col;
                const float gam = sG[cb][off] * pre;
                sG[cb][off] = gam;
                sV[cb][off] = sV[cb][off] / fmaxf(gam, 1e-30f);
            }
        }
        __syncthreads();

        // ---- P = causal_mask(Q @ K^T), [64 x 64]: 2 tiles per wave ----
        #pragma unroll
        for (int j = 0; j < 2; ++j) {
            const int T  = wid * 2 + j;
            const int ti = T >> 2, tj = T & 3;
            const int n  = lane & 15;
            const int mh = (lane >> 4) * 8;
            if (tj > ti) {
                #pragma unroll
                for (int i = 0; i < 8; ++i)
                    sP[(16 * ti + i + mh) * LSQ + 16 * tj + n] = 0.0f;
            } else {
                v8f acc = {0.f, 0.f, 0.f, 0.f, 0.f, 0.f, 0.f, 0.f};
                #pragma unroll
                for (int kk = 0; kk < Dn; kk += 4) {
                    v2f a  = frag_row<LSQ>(sQ[cb], 16 * ti, kk, lane);
                    v2f bb = frag_row<LSQ>(sK[cb], 16 * tj, kk, lane);
                    acc = wmma_f32(a, bb, acc);
                }
                #pragma unroll
                for (int i = 0; i < 8; ++i) {
                    const int m = i + mh;
                    float val = acc[i];
                    if (16 * tj + n > 16 * ti + m) val = 0.0f;  // keep s <= t
                    sP[(16 * ti + m) * LSQ + 16 * tj + n] = val;
                }
            }
        }
        __syncthreads();

        // ---- Y = Gamma ⊙ (P @ Vhat + Q @ S0), [64 x 32]: 1 tile per wave ----
        {
            const int ti = wid >> 1, tj = wid & 1;
            v8f acc = {0.f, 0.f, 0.f, 0.f, 0.f, 0.f, 0.f, 0.f};
            #pragma unroll
            for (int kk = 0; kk < Cn; kk += 4) {   // intra-chunk
                v2f a  = frag_row<LSQ>(sP, 16 * ti, kk, lane);
                v2f bb = frag_col<LSV>(sV[cb], kk, 16 * tj, lane);
                acc = wmma_f32(a, bb, acc);
            }
            #pragma unroll
            for (int kk = 0; kk < Dn; kk += 4) {   // cross-chunk (old state)
                v2f a  = frag_row<LSQ>(sQ[cb], 16 * ti, kk, lane);
                v2f bb = frag_col<LSV>(sS, kk, 16 * tj, lane);
                acc = wmma_f32(a, bb, acc);
            }
            const int n  = lane & 15;
            const int mh = (lane >> 4) * 8;
            #pragma unroll
            for (int i = 0; i < 8; ++i) {
                const int trow = 16 * ti + i + mh;
                const int vcol = 16 * tj + n;
                const float y  = acc[i] * sG[cb][trow * LSV + vcol];
                out[base + (size_t)(t0 + trow) * rowstride + (size_t)(vbase + vcol)] = y;
            }
        }
        __syncthreads();

        // ---- S0 = Gamma_last ⊙ (S0 + K^T @ Vhat), [64 x 32]: 1 tile per wave ----
        {
            const int ti = wid >> 1, tj = wid & 1;   // ti: k-dim tile, tj: v-dim tile
            v8f acc = {0.f, 0.f, 0.f, 0.f, 0.f, 0.f, 0.f, 0.f};
            #pragma unroll
            for (int kk = 0; kk < Cn; kk += 4) {     // reduce over time s
                v2f a  = frag_col<LSQ>(sK[cb], kk, 16 * ti, lane);  // (K^T)[k][s]
                v2f bb = frag_col<LSV>(sV[cb], kk, 16 * tj, lane);  // Vhat[s][v]
                acc = wmma_f32(a, bb, acc);
            }
            const int n  = lane & 15;
            const int mh = (lane >> 4) * 8;
            #pragma unroll
            for (int i = 0; i < 8; ++i) {
                const int kr   = 16 * ti + i + mh;
                const int vcol = 16 * tj + n;
                const float gl = sG[cb][(Cn - 1) * LSV + vcol];
                sS[kr * LSV + vcol] = gl * (sS[kr * LSV + vcol] + acc[i]);
            }
        }
    }
}

extern "C" void kernel_launch(void* const* d_in, const int* in_sizes, int n_in,
                              void* d_out, int out_size, void* d_ws, size_t ws_size,
                              hipStream_t stream) {
    const float* q = (const float*)d_in[0];
    const float* k = (const float*)d_in[1];
    const float* v = (const float*)d_in[2];
    const float* g = (const float*)d_in[3];
    float* out = (float*)d_out;
    dim3 grid(Bn * Hn * 2);    // 256 workgroups: (b,h) x value-half
    dim3 block(NTHREADS);      // 8 wave32s
    gla_chunk_wmma<<<grid, block, 0, stream>>>(q, k, v, g, out);
}